// GCN_60610578482005
// MI455X (gfx1250) — compile-verified
//
#include <hip/hip_runtime.h>
#include <hip/hip_bf16.h>

typedef __attribute__((ext_vector_type(2))) float v2f;
typedef __attribute__((ext_vector_type(8))) float v8f;

#define N_IN_F 128

__device__ __forceinline__ void atomic_add_f32(float* p, float v) {
    __hip_atomic_fetch_add(p, v, __ATOMIC_RELAXED, __HIP_MEMORY_SCOPE_AGENT);
}

// ---------------- zero fill (float4 vectorized, grid-stride) ----------------
__global__ void gcn_zero_v4(float4* __restrict__ p, long n4) {
    long i = (long)blockIdx.x * blockDim.x + threadIdx.x;
    long stride = (long)gridDim.x * blockDim.x;
    float4 z = make_float4(0.f, 0.f, 0.f, 0.f);
    for (; i < n4; i += stride) p[i] = z;
}

// ---------------- degree histogram via L2 float atomics ----------------
__global__ void gcn_degrees(const int* __restrict__ esrc, const int* __restrict__ edst,
                            float* __restrict__ deg_out, float* __restrict__ deg_in,
                            int n_edges) {
    int e = blockIdx.x * blockDim.x + threadIdx.x;
    if (e >= n_edges) return;
    atomic_add_f32(&deg_out[esrc[e]], 1.0f);
    atomic_add_f32(&deg_in[edst[e]], 1.0f);
}

// deg -> norm in place: deg>0 ? rsqrt(deg) : 0   (deg is an integer count, so max(deg,1) is redundant)
__global__ void gcn_norms(float* __restrict__ d, int n) {
    int i = blockIdx.x * blockDim.x + threadIdx.x;
    if (i >= n) return;
    float v = d[i];
    d[i] = (v > 0.f) ? rsqrtf(v) : 0.f;
}

// ---------------- edge scatter: agg[dst] += h[src] * norm_src[src] ----------------
// 32 lanes per edge, 4 floats/lane (F=128). Gather is float4; scatter is 4 f32 atomics (L2-resident).
__global__ void gcn_scatter128(const float* __restrict__ h,
                               const float* __restrict__ norm_src,
                               const int* __restrict__ esrc,
                               const int* __restrict__ edst,
                               float* __restrict__ agg,
                               int n_edges) {
    long t = (long)blockIdx.x * blockDim.x + threadIdx.x;
    int e = (int)(t >> 5);
    if (e >= n_edges) return;
    int lane = (int)(t & 31);
    int s = esrc[e];
    int d = edst[e];
    float ns = norm_src[s];
    const float4* hp = (const float4*)(h + (size_t)s * N_IN_F);
    float4 v = hp[lane];
    float* ap = agg + (size_t)d * N_IN_F + lane * 4;
    atomic_add_f32(ap + 0, v.x * ns);
    atomic_add_f32(ap + 1, v.y * ns);
    atomic_add_f32(ap + 2, v.z * ns);
    atomic_add_f32(ap + 3, v.w * ns);
}

// ---------------- fused (agg * norm_dst) @ W + b (+ReLU) via V_WMMA_F32_16X16X4_F32 ----------------
// Block = 256 threads = 8 waves; each wave owns a 16-row tile (block covers 128 rows).
// W [IN_F x OUT_F] staged in LDS once per block. A rows (scaled) held in 64 VGPRs per wave,
// reused across all OUT_F/16 column tiles. No divergence before WMMA (EXEC all-1s); row
// clamping is a select, store masking happens after the WMMAs.
template <int IN_F, int OUT_F, bool RELU>
__global__ void gcn_gemm_wmma(const float* __restrict__ agg,
                              const float* __restrict__ norm_dst,
                              const float* __restrict__ Wg,
                              const float* __restrict__ bias,
                              float* __restrict__ out,
                              int n_nodes) {
    __shared__ float Ws[IN_F * OUT_F];
    for (int i = threadIdx.x; i < IN_F * OUT_F; i += blockDim.x) Ws[i] = Wg[i];
    __syncthreads();

    const int lane = threadIdx.x & 31;
    const int wave = threadIdx.x >> 5;               // 0..7
    const int row0 = (blockIdx.x * 8 + wave) * 16;   // wave's 16-row tile
    const int half = lane >> 4;                      // 0: lanes 0-15, 1: lanes 16-31
    const int m    = lane & 15;

    // A-matrix load row (clamped for OOB tiles; stores masked later)
    int rowA  = row0 + m;
    int rowAc = (rowA < n_nodes) ? rowA : (n_nodes - 1);
    float nd  = norm_dst[rowAc];
    const float* arow = agg + (size_t)rowAc * IN_F;

    // A tile in registers per ISA layout: v0 = {K0 | K2}, v1 = {K1 | K3} per k-quad.
    v2f a[IN_F / 4];
#pragma unroll
    for (int kk = 0; kk < IN_F / 4; ++kk) {
        const float2 t = *(const float2*)(arow + kk * 4 + 2 * half);
        v2f av; av.x = t.x * nd; av.y = t.y * nd;
        a[kk] = av;
    }

#pragma unroll
    for (int nt = 0; nt < OUT_F / 16; ++nt) {
        const int n0 = nt * 16;
        v8f acc = {};
#pragma unroll
        for (int kk = 0; kk < IN_F / 4; ++kk) {
            const int kb = kk * 4 + 2 * half;        // B rows mirror A's half-split
            v2f b;
            b.x = Ws[(kb + 0) * OUT_F + n0 + m];
            b.y = Ws[(kb + 1) * OUT_F + n0 + m];
            acc = __builtin_amdgcn_wmma_f32_16x16x4_f32(
                false, a[kk], false, b, (short)0, acc, false, false);
        }
        // C/D layout: VGPR r -> row (half*8 + r), col n0 + m
        const int col = n0 + m;
        const float bv = bias[col];
#pragma unroll
        for (int r = 0; r < 8; ++r) {
            const int rr = row0 + half * 8 + r;
            if (rr < n_nodes) {
                float v = acc[r] + bv;
                if (RELU) v = fmaxf(v, 0.f);
                out[(size_t)rr * OUT_F + col] = v;
            }
        }
    }
}

extern "C" void kernel_launch(void* const* d_in, const int* in_sizes, int n_in,
                              void* d_out, int out_size, void* d_ws, size_t ws_size,
                              hipStream_t stream) {
    const float* x    = (const float*)d_in[0];
    const int*   esrc = (const int*)d_in[1];
    const int*   edst = (const int*)d_in[2];
    const float* W0   = (const float*)d_in[3];
    const float* b0   = (const float*)d_in[4];
    const float* W1   = (const float*)d_in[5];
    const float* b1   = (const float*)d_in[6];
    const float* W2   = (const float*)d_in[7];
    const float* b2   = (const float*)d_in[8];
    float* out = (float*)d_out;

    const int N = in_sizes[0] / N_IN_F;   // 100000
    const int E = in_sizes[1];            // 1600000

    // workspace layout (floats): [norm_src N][norm_dst N][agg N*128]
    float* nsrc = (float*)d_ws;
    float* ndst = nsrc + N;
    float* agg  = ndst + N;

    const long out_h1 = (long)N * 128;
    float* h1 = out + out_h1;             // layer-0 output
    float* h2 = out + 2 * out_h1;         // layer-1 output
    float* h3 = out + 3 * out_h1;         // layer-2 output (64-wide)

    // outputs[0] = x (straight copy)
    hipMemcpyAsync(out, x, (size_t)N * N_IN_F * sizeof(float),
                   hipMemcpyDeviceToDevice, stream);

    // degrees -> norms
    {
        long n4 = (long)(2 * N) / 4;
        gcn_zero_v4<<<(int)((n4 + 255) / 256), 256, 0, stream>>>((float4*)nsrc, n4);
        gcn_degrees<<<(E + 255) / 256, 256, 0, stream>>>(esrc, edst, nsrc, ndst, E);
        gcn_norms<<<(2 * N + 255) / 256, 256, 0, stream>>>(nsrc, 2 * N);
    }

    const long aggN4   = (long)N * N_IN_F / 4;
    const int  zeroGrd = (int)((aggN4 + 255) / 256) < 16384 ? (int)((aggN4 + 255) / 256) : 16384;
    const int  scatGrd = (int)(((long)E * 32 + 255) / 256);
    const int  gemmGrd = (N + 127) / 128;

    // ---- layer 0: x -> h1 (128->128, ReLU) ----
    gcn_zero_v4<<<zeroGrd, 256, 0, stream>>>((float4*)agg, aggN4);
    gcn_scatter128<<<scatGrd, 256, 0, stream>>>(x, nsrc, esrc, edst, agg, E);
    gcn_gemm_wmma<128, 128, true><<<gemmGrd, 256, 0, stream>>>(agg, ndst, W0, b0, h1, N);

    // ---- layer 1: h1 -> h2 (128->128, ReLU) ----
    gcn_zero_v4<<<zeroGrd, 256, 0, stream>>>((float4*)agg, aggN4);
    gcn_scatter128<<<scatGrd, 256, 0, stream>>>(h1, nsrc, esrc, edst, agg, E);
    gcn_gemm_wmma<128, 128, true><<<gemmGrd, 256, 0, stream>>>(agg, ndst, W1, b1, h2, N);

    // ---- layer 2: h2 -> h3 (128->64, no ReLU) ----
    gcn_zero_v4<<<zeroGrd, 256, 0, stream>>>((float4*)agg, aggN4);
    gcn_scatter128<<<scatGrd, 256, 0, stream>>>(h2, nsrc, esrc, edst, agg, E);
    gcn_gemm_wmma<128, 64, false><<<gemmGrd, 256, 0, stream>>>(agg, ndst, W2, b2, h3, N);
}